// Graph_Encoder_43026982371787
// MI455X (gfx1250) — compile-verified
//
#include <hip/hip_runtime.h>
#include <hip/hip_bf16.h>

typedef __attribute__((ext_vector_type(2))) float v2f;
typedef __attribute__((ext_vector_type(8))) float v8f;

#define D_IN 256
#define D_OUT 64
// B-fragment table: 64 K-quads * 4 N-tiles * 32 lanes float2 = 64 KB LDS
#define NFRAG (64 * 4 * 32)

// ---------------------------------------------------------------------------
// 1) zero output accumulator, set deg = 1.0 (self loop)
// ---------------------------------------------------------------------------
__global__ void gcn_init_kernel(float* __restrict__ out, float* __restrict__ deg,
                                int n, int tot) {
    int gid = blockIdx.x * blockDim.x + threadIdx.x;
    if (gid < tot) out[gid] = 0.0f;
    if (gid < n)   deg[gid] = 1.0f;   // self-loop contributes 1 to each degree
}

// ---------------------------------------------------------------------------
// 2) degree over destination nodes
// ---------------------------------------------------------------------------
__global__ void gcn_degree_kernel(const int* __restrict__ col, float* __restrict__ deg,
                                  int nE) {
    int e = blockIdx.x * blockDim.x + threadIdx.x;
    if (e >= nE) return;
    atomicAdd(&deg[col[e]], 1.0f);
}

// ---------------------------------------------------------------------------
// 3) deg -> rsqrt(deg) in place
// ---------------------------------------------------------------------------
__global__ void gcn_rsqrt_kernel(float* __restrict__ deg, int n) {
    int i = blockIdx.x * blockDim.x + threadIdx.x;
    if (i >= n) return;
    float d = deg[i];
    deg[i] = (d > 0.0f) ? rsqrtf(d) : 0.0f;
}

// ---------------------------------------------------------------------------
// 4) h = x @ W with V_WMMA_F32_16X16X4_F32, W staged in LDS in fragment order.
//
// ISA layouts (CDNA5 ISA):
//   A 16x4 f32 : lanes 0-15 -> M=0..15, K pair {0,1}; lanes 16-31 -> K pair {2,3}
//   B 4x16 f32 : lanes 0-15 -> N=0..15, rows {K=0,K=1}; lanes 16-31 -> rows {K=2,K=3}
//   C/D 16x16  : lanes 0-15: VGPR v=(M=v,N=lane); lanes 16-31: (M=8+v,N=lane-16)
//
// LDS fragment table: wfrag[(q*4 + j)*32 + lane] = {W[4q+2h][16j+l], W[4q+2h+1][16j+l]}
// where h=lane>>4, l=lane&15. A wave's fragment read is 256 contiguous bytes
// -> all 64 LDS banks touched once (conflict-free ds_load_b64).
// ---------------------------------------------------------------------------
__global__ void gcn_gemm_wmma_kernel(const float* __restrict__ x,
                                     const float* __restrict__ W,
                                     float* __restrict__ h, int n) {
    extern __shared__ float2 wfrag[];   // 8192 float2 = 64 KB (WGP LDS is 320 KB)

    int tid  = threadIdx.x;
    int lane = tid & 31;
    int half = lane >> 4;               // 0: K pair {0,1}; 1: K pair {2,3}
    int l    = lane & 15;               // M index for A, N index for B/C

    // --- cooperative staging of W into fragment order (once per block) ---
    for (int i = tid; i < NFRAG; i += 256) {
        int lane2 = i & 31;
        int j2    = (i >> 5) & 3;
        int q2    = i >> 7;
        int h2    = lane2 >> 4;
        int l2    = lane2 & 15;
        int r0    = 4 * q2 + 2 * h2;
        float2 v;
        v.x = W[(size_t)r0 * D_OUT + j2 * 16 + l2];
        v.y = W[(size_t)(r0 + 1) * D_OUT + j2 * 16 + l2];
        wfrag[i] = v;
    }
    __syncthreads();                    // barrier before any wave may exit

    int wave = (blockIdx.x * blockDim.x + tid) >> 5;
    int m0 = wave * 16;
    if (m0 >= n) return;                // wave-uniform: live waves keep EXEC all-1s

    v8f acc0 = {}, acc1 = {}, acc2 = {}, acc3 = {};

    const float* xrow = x + (size_t)(m0 + l) * D_IN;

#pragma unroll 4
    for (int k0 = 0; k0 < D_IN; k0 += 4) {
        int kk = k0 + 2 * half;
        v2f a;
        a.x = xrow[kk + 0];
        a.y = xrow[kk + 1];

        const float2* bf = &wfrag[(size_t)(k0 >> 2) * 128 + lane];
        float2 f0 = bf[0];              // j = 0
        float2 f1 = bf[32];             // j = 1
        float2 f2 = bf[64];             // j = 2
        float2 f3 = bf[96];             // j = 3
        v2f b0 = {f0.x, f0.y};
        v2f b1 = {f1.x, f1.y};
        v2f b2 = {f2.x, f2.y};
        v2f b3 = {f3.x, f3.y};

        acc0 = __builtin_amdgcn_wmma_f32_16x16x4_f32(false, a, false, b0, (short)0, acc0, false, false);
        acc1 = __builtin_amdgcn_wmma_f32_16x16x4_f32(false, a, false, b1, (short)0, acc1, false, false);
        acc2 = __builtin_amdgcn_wmma_f32_16x16x4_f32(false, a, false, b2, (short)0, acc2, false, false);
        acc3 = __builtin_amdgcn_wmma_f32_16x16x4_f32(false, a, false, b3, (short)0, acc3, false, false);
    }

    // store: lane (half,l), VGPR v  ->  h[m0 + 8*half + v][jtile*16 + l]
#pragma unroll
    for (int v = 0; v < 8; ++v) {
        size_t r = (size_t)(m0 + 8 * half + v) * D_OUT;
        h[r + l +  0] = acc0[v];
        h[r + l + 16] = acc1[v];
        h[r + l + 32] = acc2[v];
        h[r + l + 48] = acc3[v];
    }
}

// ---------------------------------------------------------------------------
// 5) edge scatter: one wave per edge; lane i handles dims {2i, 2i+1}.
//    norm = dinv[row]*dinv[col];  out[col] += h[row]*norm  (atomic f32, L2-resident)
// ---------------------------------------------------------------------------
__global__ void gcn_scatter_kernel(const int* __restrict__ row,
                                   const int* __restrict__ col,
                                   const float* __restrict__ h,
                                   const float* __restrict__ dinv,
                                   float* __restrict__ out, int nE) {
    long long gid = (long long)blockIdx.x * blockDim.x + threadIdx.x;
    int e    = (int)(gid >> 5);
    int lane = (int)(gid & 31);
    if (e >= nE) return;
    int r = row[e];                       // wave-uniform broadcast load
    int c = col[e];
    float nrm = dinv[r] * dinv[c];
    const float2* hs = (const float2*)(h + (size_t)r * D_OUT);
    float2 v = hs[lane];
    float* o = out + (size_t)c * D_OUT + lane * 2;
    atomicAdd(o + 0, v.x * nrm);
    atomicAdd(o + 1, v.y * nrm);
}

// ---------------------------------------------------------------------------
// 6) self-loop term + bias + ReLU (in place on out)
// ---------------------------------------------------------------------------
__global__ void gcn_finalize_kernel(const float* __restrict__ h,
                                    const float* __restrict__ dinv,
                                    const float* __restrict__ b,
                                    float* __restrict__ out, int tot) {
    int gid = blockIdx.x * blockDim.x + threadIdx.x;
    if (gid >= tot) return;
    int i = gid >> 6;        // node
    int d = gid & 63;        // feature
    float di = dinv[i];
    float v = out[gid] + h[gid] * (di * di) + b[d];
    out[gid] = fmaxf(v, 0.0f);
}

// ---------------------------------------------------------------------------
// launcher
// ---------------------------------------------------------------------------
extern "C" void kernel_launch(void* const* d_in, const int* in_sizes, int n_in,
                              void* d_out, int out_size, void* d_ws, size_t ws_size,
                              hipStream_t stream) {
    const float* x  = (const float*)d_in[0];     // [N, 256]
    const int*   ei = (const int*)  d_in[1];     // [2, E] flat: row ptr, then col ptr
    const float* W  = (const float*)d_in[2];     // [256, 64]
    const float* b  = (const float*)d_in[3];     // [64]
    float* out = (float*)d_out;                  // [N, 64]

    int n = in_sizes[0] / D_IN;                  // 100000
    int E = in_sizes[1] / 2;                     // 1600000
    const int* row = ei;
    const int* col = ei + E;

    // workspace: h [N*64] f32, then dinv/deg [N] f32  (~26 MB)
    float* h    = (float*)d_ws;
    float* dinv = h + (size_t)n * D_OUT;

    int tot = n * D_OUT;

    gcn_init_kernel<<<(tot + 255) / 256, 256, 0, stream>>>(out, dinv, n, tot);
    gcn_degree_kernel<<<(E + 255) / 256, 256, 0, stream>>>(col, dinv, E);
    gcn_rsqrt_kernel<<<(n + 255) / 256, 256, 0, stream>>>(dinv, n);

    int mtiles = (n + 15) / 16;                  // 6250 waves, 8 waves per 256-thread block
    size_t lds_bytes = (size_t)NFRAG * sizeof(float2);   // 64 KB fragment table
    gcn_gemm_wmma_kernel<<<(mtiles + 7) / 8, 256, lds_bytes, stream>>>(x, W, h, n);

    long long sthreads = (long long)E * 32;      // one wave per edge
    gcn_scatter_kernel<<<(unsigned)((sthreads + 255) / 256), 256, 0, stream>>>(
        row, col, h, dinv, out, E);

    gcn_finalize_kernel<<<(tot + 255) / 256, 256, 0, stream>>>(h, dinv, b, out, tot);
}